// TCAttention_5308579577938
// MI455X (gfx1250) — compile-verified
//
#include <hip/hip_runtime.h>
#include <hip/hip_bf16.h>

typedef __attribute__((ext_vector_type(2))) float v2f;
typedef __attribute__((ext_vector_type(8))) float v8f;
typedef int v4i __attribute__((vector_size(16)));

#define BATCH 4
#define TNUM 20
#define N_TOK 197
#define CDIM 768
#define HNUM 12
#define HD 64
#define QKVC 2304
#define MROWS 15760      // B*N = 80*197
#define NPAD 208         // 13*16
#define NPANELS 13
#define KB 32            // K-chunk staged in LDS per GEMM block
#define LAPITCH 36       // padded LDS row pitch (16B-aligned, conflict-free)

__device__ __forceinline__ v8f wmma4(v2f a, v2f b, v8f c) {
    // D(16x16,f32) = A(16x4,f32) x B(4x16,f32) + C
    return __builtin_amdgcn_wmma_f32_16x16x4_f32(
        false, a, false, b, (short)0, c, false, false);
}

#if defined(__HIP_DEVICE_COMPILE__) && \
    __has_builtin(__builtin_amdgcn_global_load_async_to_lds_b128) && \
    __has_builtin(__builtin_amdgcn_s_wait_asynccnt)
#define HAVE_ASYNC_LDS 1
#else
#define HAVE_ASYNC_LDS 0
#if defined(__HIP_DEVICE_COMPILE__)
#warning "gfx1250 async-to-LDS builtins not found; using VALU staging fallback"
#endif
#endif

#if HAVE_ASYNC_LDS
typedef __attribute__((address_space(1))) v4i* gas_v4i_ptr;
typedef __attribute__((address_space(3))) v4i* las_v4i_ptr;
__device__ __forceinline__ void async_copy16(const float* g, float* l) {
    // GLOBAL_LOAD_ASYNC_TO_LDS_B128 (ASYNCcnt); AS1 src, AS3 dst.
    // generic->local addrspacecast on AMDGPU == truncate to low 32 bits.
    __builtin_amdgcn_global_load_async_to_lds_b128(
        (gas_v4i_ptr)(uintptr_t)g,
        (las_v4i_ptr)(uint32_t)(uintptr_t)l,
        0, 0);
}
#endif

// ---------------------------------------------------------------------------
// Wt[k][n] = W[n][k];  W is (R rows x Cc cols) row-major, Wt is (Cc x R)
__global__ void transpose_kernel(const float* __restrict__ W,
                                 float* __restrict__ Wt, int R, int Cc) {
    int idx = blockIdx.x * blockDim.x + threadIdx.x;
    if (idx < R * Cc) {
        int r = idx / Cc, c = idx % Cc;
        Wt[(size_t)c * R + r] = W[(size_t)r * Cc + c];
    }
}

// ---------------------------------------------------------------------------
// C[M,N] = A[M,K] @ Bt[K,N] (+ bias[N]).  Block = 128 thr (4 waves).
// Block tile: 64 (M) x 256 (N); each wave: 64 x 64 (4x4 accumulators).
// A chunk (64 x KB) is async-staged into LDS once per block and shared by
// all 4 waves (4x less A traffic); B read direct (L2-resident).
// Requires N%256==0, K%KB==0; M ragged edge handled by clamp+guard.
__global__ __launch_bounds__(128) void gemm_wmma(
    const float* __restrict__ A, const float* __restrict__ Bt,
    float* __restrict__ C, int M, int N, int K, const float* __restrict__ bias)
{
    __shared__ float LA[64][LAPITCH];   // 9.2 KB staged A chunk

    int tid = threadIdx.x;
    int lane = tid & 31;
    int wave = tid >> 5;
    int l16 = lane & 15;
    int lhalf = (lane >= 16) ? 1 : 0;

    int rbase = blockIdx.x * 64;
    int cbase = blockIdx.y * 256 + wave * 64;

    v8f zero = {};
    v8f acc[4][4];
#pragma unroll
    for (int m = 0; m < 4; ++m)
#pragma unroll
        for (int s = 0; s < 4; ++s) acc[m][s] = zero;

    // Loader mapping: 2 threads per A row, 64B (16 floats) each.
    int lrow = tid >> 1;
    int lcol = (tid & 1) * (KB / 2);
    int grow = rbase + lrow;
    if (grow > M - 1) grow = M - 1;          // clamp ragged M edge
    const float* gsrc_row = A + (size_t)grow * K + lcol;

    for (int kc = 0; kc < K; kc += KB) {
        // ---- stage A chunk (64 x KB) into LDS ----
#if HAVE_ASYNC_LDS
#pragma unroll
        for (int q = 0; q < (KB / 2) / 4; ++q)
            async_copy16(gsrc_row + kc + q * 4, &LA[lrow][lcol + q * 4]);
        __builtin_amdgcn_s_wait_asynccnt(0);
#else
#pragma unroll
        for (int q = 0; q < (KB / 2) / 4; ++q) {
            float4 vv = *(const float4*)(gsrc_row + kc + q * 4);
            *(float4*)&LA[lrow][lcol + q * 4] = vv;
        }
#endif
        __syncthreads();

        // ---- compute KB/4 K-steps ----
#pragma unroll
        for (int ks = 0; ks < KB; ks += 4) {
            int ka = ks + lhalf * 2;
            v2f af[4];
#pragma unroll
            for (int m = 0; m < 4; ++m) {
                v2f a = {LA[m * 16 + l16][ka], LA[m * 16 + l16][ka + 1]};
                af[m] = a;
            }
            int kaG = kc + ka;
            const float* b0 = Bt + (size_t)kaG * N + cbase + l16;
            const float* b1 = b0 + N;
#pragma unroll
            for (int s = 0; s < 4; ++s) {
                v2f bf = {b0[s * 16], b1[s * 16]};
#pragma unroll
                for (int m = 0; m < 4; ++m)
                    acc[m][s] = wmma4(af[m], bf, acc[m][s]);
            }
        }
        __syncthreads();
    }

#pragma unroll
    for (int s = 0; s < 4; ++s) {
        int cn = cbase + s * 16 + l16;
        float bv = bias ? bias[cn] : 0.0f;
#pragma unroll
        for (int m = 0; m < 4; ++m)
#pragma unroll
            for (int r = 0; r < 8; ++r) {
                int rr = rbase + m * 16 + r + lhalf * 8;
                if (rr < M) C[(size_t)rr * N + cn] = acc[m][s][r] + bv;
            }
    }
}

// ---------------------------------------------------------------------------
// In-place weighted prefix sum over T for the k and v planes (cols 768..2303).
// k_new[t] = sum_{s<=t} w[s]*k[s]  (forward-pass simplification of the ref)
__global__ void cumsum_kv(float* __restrict__ qkv, const float* __restrict__ w) {
    int idx = blockIdx.x * blockDim.x + threadIdx.x;
    const int total = BATCH * N_TOK * 1536;
    if (idx >= total) return;
    int cc = idx % 1536;
    int rest = idx / 1536;
    int n = rest % N_TOK;
    int b = rest / N_TOK;
    int col = CDIM + cc;           // k plane then v plane
    float acc = 0.0f;
    for (int t = 0; t < TNUM; ++t) {
        size_t off = ((size_t)(b * TNUM + t) * N_TOK + n) * QKVC + col;
        acc += w[t] * qkv[off];
        qkv[off] = acc;
    }
}

// ---------------------------------------------------------------------------
// Attention: one block per (head, 16-row panel).  grid = 960*13 = 12480.
// Phase 1: S = scale*q@k^T (WMMA), blend with attn_res -> attn_out + LDS.
// Phase 2: masked softmax with masked-entries-as-exp(0) semantics (in LDS).
// Phase 3: out = P @ v (WMMA) -> headflat in (b,H,t,n,d) order (ref transpose).
__global__ __launch_bounds__(128) void attn_kernel(
    const float* __restrict__ qkv, const float* __restrict__ attn_res,
    const float* __restrict__ alpha, float* __restrict__ attn_out,
    float* __restrict__ headflat)
{
    __shared__ float Sp[16][NPAD];   // 13.3 KB: S panel, then P panel in place

    int blk = blockIdx.x;
    int ti = blk % NPANELS;          // row panel index
    int bth = blk / NPANELS;         // (b*T + t)*H + h
    int h = bth % HNUM;
    int bt = bth / HNUM;

    int tid = threadIdx.x;
    int lane = tid & 31;
    int wave = tid >> 5;
    int l16 = lane & 15;
    int lhalf = (lane >= 16) ? 1 : 0;

    float sblend = 1.0f / (1.0f + __expf(-alpha[0]));
    const float scale = 0.125f;      // hd^{-1/2} = 1/8

    const float* qbase = qkv + (size_t)bt * N_TOK * QKVC + h * HD;
    const float* kbase = qbase + CDIM;
    const float* vbase = qbase + 2 * CDIM;
    const float* arbase = attn_res + (size_t)bth * N_TOK * N_TOK;
    float* aobase = attn_out + (size_t)bth * N_TOK * N_TOK;

    int row0 = ti * 16;

    // Preload the 16 A-fragments of the q panel (hd=64 -> 16 K-steps of 4)
    v2f qa[16];
    {
        int r = row0 + l16;
        bool rvalid = (r < N_TOK);
        const float* qrow = qbase + (size_t)r * QKVC;
#pragma unroll
        for (int s = 0; s < 16; ++s) {
            int kk = s * 4 + lhalf * 2;
            v2f a = {rvalid ? qrow[kk] : 0.0f, rvalid ? qrow[kk + 1] : 0.0f};
            qa[s] = a;
        }
    }

    // ---- Phase 1: S panel (16 x 208), 13 column tiles over 4 waves ----
    for (int tj = wave; tj < NPANELS; tj += 4) {
        v8f acc = {};
        int cn = tj * 16 + l16;
        bool cvalid = (cn < N_TOK);
        const float* krow = kbase + (size_t)cn * QKVC;
#pragma unroll
        for (int s = 0; s < 16; ++s) {
            int kk = s * 4 + lhalf * 2;
            v2f bf = {cvalid ? krow[kk] : 0.0f, cvalid ? krow[kk + 1] : 0.0f};
            acc = wmma4(qa[s], bf, acc);
        }
#pragma unroll
        for (int r = 0; r < 8; ++r) {
            int mi = r + lhalf * 8;
            int gi = row0 + mi;
            int gj = tj * 16 + l16;
            if (gi < N_TOK && gj < N_TOK) {
                float sv = acc[r] * scale;
                float blended = sblend * sv +
                                (1.0f - sblend) * arbase[(size_t)gi * N_TOK + gj];
                aobase[(size_t)gi * N_TOK + gj] = blended;   // attn_res_out
                Sp[mi][gj] = blended;
            }
        }
    }
    __syncthreads();

    // ---- Phase 2: softmax over rows (masked entries contribute exp(0)) ----
    if (tid < 16) {
        int gi = row0 + tid;
        if (gi < N_TOK) {
            float m = 0.0f;                      // masked zeros are in the max
            for (int j = 0; j <= gi; ++j) m = fmaxf(m, Sp[tid][j]);
            float em = __expf(-m);
            float denom = (float)(N_TOK - 1 - gi) * em;  // masked contributions
            for (int j = 0; j <= gi; ++j) denom += __expf(Sp[tid][j] - m);
            float inv = 1.0f / denom;
            float fill = em * inv;               // prob at masked positions
            for (int j = 0; j <= gi; ++j) Sp[tid][j] = __expf(Sp[tid][j] - m) * inv;
            for (int j = gi + 1; j < N_TOK; ++j) Sp[tid][j] = fill;
            for (int j = N_TOK; j < NPAD; ++j) Sp[tid][j] = 0.0f;
        } else {
            for (int j = 0; j < NPAD; ++j) Sp[tid][j] = 0.0f;
        }
    }
    __syncthreads();

    // ---- Phase 3: out(16x64) = P(16x208) @ V(208x64); wave w -> d cols w*16 ----
    {
        int dcol = wave * 16;
        v8f acc = {};
        for (int s = 0; s < NPAD / 4; ++s) {
            int kk = s * 4 + lhalf * 2;
            v2f af = {Sp[l16][kk], Sp[l16][kk + 1]};
            float b0 = (kk < N_TOK)
                         ? vbase[(size_t)kk * QKVC + dcol + l16] : 0.0f;
            float b1 = (kk + 1 < N_TOK)
                         ? vbase[(size_t)(kk + 1) * QKVC + dcol + l16] : 0.0f;
            v2f bf = {b0, b1};
            acc = wmma4(af, bf, acc);
        }
        // Reference's transpose (0,2,1,3,4): store in (b, H, t, n, d) order
        int b = bt / TNUM, t = bt % TNUM;
        size_t hb = (((size_t)b * HNUM + h) * TNUM + t) * (size_t)N_TOK * HD;
#pragma unroll
        for (int r = 0; r < 8; ++r) {
            int mi = r + lhalf * 8;
            int gi = row0 + mi;
            if (gi < N_TOK)
                headflat[hb + (size_t)gi * HD + dcol + l16] = acc[r];
        }
    }
}

// ---------------------------------------------------------------------------
extern "C" void kernel_launch(void* const* d_in, const int* in_sizes, int n_in,
                              void* d_out, int out_size, void* d_ws, size_t ws_size,
                              hipStream_t stream) {
    const float* x        = (const float*)d_in[0];
    const float* W_qkv    = (const float*)d_in[1];
    const float* W_proj   = (const float*)d_in[2];
    const float* b_proj   = (const float*)d_in[3];
    const float* tmp_w    = (const float*)d_in[4];
    const float* alpha    = (const float*)d_in[5];
    const float* attn_res = (const float*)d_in[6];

    float* out      = (float*)d_out;                         // (15760, 768)
    float* attn_out = out + (size_t)MROWS * CDIM;            // (4,20,12,197,197)

    float* ws       = (float*)d_ws;
    float* qkv      = ws;                                    // 15760 x 2304
    float* headflat = qkv + (size_t)MROWS * QKVC;            // 15760 x 768
    float* WqkvT    = headflat + (size_t)MROWS * CDIM;       // 768 x 2304
    float* WprojT   = WqkvT + (size_t)CDIM * QKVC;           // 768 x 768

    // 1) Transpose weights for coalesced GEMM B-loads
    transpose_kernel<<<(QKVC * CDIM + 255) / 256, 256, 0, stream>>>(
        W_qkv, WqkvT, QKVC, CDIM);
    transpose_kernel<<<(CDIM * CDIM + 255) / 256, 256, 0, stream>>>(
        W_proj, WprojT, CDIM, CDIM);

    // 2) QKV GEMM: (15760x768) @ (768x2304)
    dim3 g1((MROWS + 63) / 64, QKVC / 256);
    gemm_wmma<<<g1, 128, 0, stream>>>(x, WqkvT, qkv, MROWS, QKVC, CDIM, nullptr);

    // 3) Weighted cumsum over T for k and v planes (in place)
    int ctotal = BATCH * N_TOK * 1536;
    cumsum_kv<<<(ctotal + 255) / 256, 256, 0, stream>>>(qkv, tmp_w);

    // 4) Attention: 960 heads x 13 row panels
    attn_kernel<<<960 * NPANELS, 128, 0, stream>>>(qkv, attn_res, alpha,
                                                   attn_out, headflat);

    // 5) Proj GEMM + bias: (15760x768) @ (768x768)
    dim3 g2((MROWS + 63) / 64, CDIM / 256);
    gemm_wmma<<<g2, 128, 0, stream>>>(headflat, WprojT, out, MROWS, CDIM, CDIM,
                                      b_proj);
}